// InjectedEncoder_64115271794967
// MI455X (gfx1250) — compile-verified
//
#include <hip/hip_runtime.h>
#include <hip/hip_bf16.h>
#include <math.h>

// ---------------------------------------------------------------------------
// MI455X (gfx1250) implementation of the InjectedEncoder forward pass.
// All heavy GEMMs run on the matrix pipe via V_WMMA_F32_16X16X4_F32 (exact
// fp32 accumulate). One wave owns one 16x16 output tile.
// - gemm16: shared A operand staged through LDS, branch-free WMMA loop.
// - dyna_gemm: templated on (K, HW) so the K-loop fully unrolls and every
//   load/store uses immediate offsets off a single base address.
// ---------------------------------------------------------------------------

typedef __attribute__((ext_vector_type(2))) float v2f;
typedef __attribute__((ext_vector_type(8))) float v8f;

#define Bsz   16
#define HW0   4096
#define T_FF  16640
#define T_FO  12480
#define T_CC  20736
#define CSSc  8353

#define G16_CHUNK   128
#define G16_STRIDE  130   // LDS row stride (floats): rows land on distinct banks

__device__ __forceinline__ v8f wmma4(v2f a, v2f b, v8f c) {
  return __builtin_amdgcn_wmma_f32_16x16x4_f32(false, a, false, b, (short)0, c,
                                               false, false);
}

// ----------------------------- WMMA GEMM kernels ---------------------------

// C[16,N] = A[16,K] @ W[K,N] + bias[N].
// Block = 256 threads = 8 waves; each wave owns one 16-col tile.
// A chunks are cooperatively staged into LDS with zero padding past K, so the
// WMMA loop is completely branch-free; B row indices are clamped (A==0 there).
__global__ void __launch_bounds__(256) gemm16_wmma(
    const float* __restrict__ A, const float* __restrict__ W,
    const float* __restrict__ bias, float* __restrict__ C, int K, int N) {
  __shared__ float Alds[16 * G16_STRIDE];
  int tid = threadIdx.x;
  int lane = tid & 31;
  int gwave = blockIdx.x * 8 + (tid >> 5);
  int ntiles = N >> 4;
  int tile = (gwave < ntiles) ? gwave : (ntiles - 1);  // clamp: dup work, same vals
  int n0 = tile << 4;
  int lh = lane & 15, lx = lane >> 4;
  int col = n0 + lh;
  v8f acc = {};
  for (int k0 = 0; k0 < K; k0 += G16_CHUNK) {
    __syncthreads();
    // stage A[0:16, k0:k0+128] -> LDS, zero-padded (clamped addr + select)
#pragma unroll
    for (int i = 0; i < 8; i++) {
      int e = tid + i * 256;          // 2048 elements
      int row = e >> 7, kk = e & 127;
      int kg = k0 + kk;
      int kc = (kg < K) ? kg : (K - 1);
      float v = A[(size_t)row * K + kc];
      Alds[row * G16_STRIDE + kk] = (kg < K) ? v : 0.f;
    }
    __syncthreads();
    const float* arow = &Alds[lh * G16_STRIDE];
#pragma unroll 4
    for (int kk = 0; kk < G16_CHUNK; kk += 4) {
      int ka = kk + (lx << 1);
      v2f a = *(const v2f*)(arow + ka);              // ds_load_b64
      int kg0 = k0 + ka;
      int kg1 = kg0 + 1;
      kg0 = (kg0 < K) ? kg0 : (K - 1);               // clamp; A pad is zero
      kg1 = (kg1 < K) ? kg1 : (K - 1);
      v2f b;
      b.x = W[(size_t)kg0 * N + col];
      b.y = W[(size_t)kg1 * N + col];
      acc = wmma4(a, b, acc);
    }
  }
  float bv = bias ? bias[col] : 0.f;
#pragma unroll
  for (int r = 0; r < 8; r++) {
    int ro = r + (lx << 3);           // C layout: lanes>=16 hold M=r+8
    C[(size_t)ro * N + col] = acc[r] + bv;
  }
}

// Batched hyper-network GEMM (compile-time KD, HW): for each sample b,
//   D[b] = ksW[b] ([M,KD] row-major) @ X[b] ([KD,HW]) + ksBias[b][M]
// optional elementwise addend (shortcut) and optional ReLU.
// Fully unrolled K loop; all loads/stores use immediate offsets.
template <int KD, int HWc>
__global__ void __launch_bounds__(256) dyna_gemm_t(
    const float* __restrict__ ks, int kstride, int woff, int boff,
    const float* __restrict__ X, float* __restrict__ D,
    const float* __restrict__ add, int M, int relu) {
  int gtid = blockIdx.x * blockDim.x + threadIdx.x;
  int wave = gtid >> 5;
  int lane = gtid & 31;
  const int nt = HWc >> 4;
  int mt = M >> 4;
  int wpb = mt * nt;
  if (wave >= Bsz * wpb) return;          // wave-uniform exit
  int b = wave / wpb;
  int r2 = wave % wpb;
  int m0 = (r2 / nt) << 4;
  int n0 = (r2 % nt) << 4;
  int lh = lane & 15, lx = lane >> 4;
  int col = n0 + lh;
  // single base address per operand; loop offsets are compile-time constants
  const float* Ap = ks + (size_t)b * kstride + woff + (size_t)(m0 + lh) * KD +
                    (lx << 1);
  const float* Bp = X + (size_t)b * KD * HWc + (size_t)(lx << 1) * HWc + col;
  const float* bv = ks + (size_t)b * kstride + boff;
  v8f acc = {};
#pragma unroll
  for (int k = 0; k < KD; k += 4) {
    v2f a = *(const v2f*)(Ap + k);                  // imm-offset b64 load
    v2f bb;
    bb.x = Bp[(size_t)k * HWc];                     // imm-offset b32 loads
    bb.y = Bp[(size_t)(k + 1) * HWc];
    acc = wmma4(a, bb, acc);
  }
  float* Dp = D + (size_t)b * M * HWc + (size_t)(m0 + (lx << 3)) * HWc + col;
  const float* Sp =
      add ? add + (size_t)b * M * HWc + (size_t)(m0 + (lx << 3)) * HWc + col
          : nullptr;
#pragma unroll
  for (int r = 0; r < 8; r++) {
    float v = acc[r] + bv[m0 + r + (lx << 3)];
    if (Sp) v += Sp[(size_t)r * HWc];
    if (relu) v = fmaxf(v, 0.f);
    Dp[(size_t)r * HWc] = v;
  }
}

// ----------------------------- support kernels -----------------------------

__global__ void k_gather_labs(const float* __restrict__ W,
                              const float* __restrict__ bvec,
                              const int* __restrict__ labels,
                              float* __restrict__ out) {
  int tid = blockIdx.x * blockDim.x + threadIdx.x;
  if (tid >= Bsz * 256) return;
  int b = tid >> 8, j = tid & 255;
  out[tid] = W[(size_t)labels[b] * 256 + j] + bvec[j];
}

__global__ void k_conv3x3(const float* __restrict__ in,
                          const float* __restrict__ w,
                          const float* __restrict__ bias,
                          float* __restrict__ out, int B, int Cin, int Cout) {
  int tid = blockIdx.x * blockDim.x + threadIdx.x;
  if (tid >= B * Cout * HW0) return;
  int p = tid & 4095;
  int rest = tid >> 12;
  int co = rest % Cout;
  int b = rest / Cout;
  int y = p >> 6, x = p & 63;
  float acc = bias[co];
  const float* wb = w + (size_t)co * Cin * 9;
  const float* ib = in + (size_t)b * Cin * HW0;
  for (int ci = 0; ci < Cin; ci++) {
    const float* ic = ib + (size_t)ci * HW0;
    const float* wc = wb + ci * 9;
#pragma unroll
    for (int dy = -1; dy <= 1; dy++) {
      int yy = y + dy;
      if ((unsigned)yy >= 64u) continue;
#pragma unroll
      for (int dx = -1; dx <= 1; dx++) {
        int xx = x + dx;
        if ((unsigned)xx >= 64u) continue;
        acc += ic[yy * 64 + xx] * wc[(dy + 1) * 3 + (dx + 1)];
      }
    }
  }
  out[tid] = acc;
}

// multi_cut on a [B,224,64,64] tensor -> css [B,8353]
__global__ void k_multicut(const float* __restrict__ cv,
                           float* __restrict__ css) {
  int tid = blockIdx.x * blockDim.x + threadIdx.x;
  if (tid >= Bsz * CSSc) return;
  int b = tid / CSSc, j = tid % CSSc;
  const float* base = cv + (size_t)b * 224 * HW0;
  float acc = 0.f;
  if (j < 32) {                               // c0: mean over H,W
    const float* p = base + (size_t)j * HW0;
    for (int i = 0; i < 4096; i++) acc += p[i];
    acc *= (1.f / 4096.f);
  } else if (j < 96) {                        // c1: per-H, mean over ch,W
    int h = j - 32;
    for (int c = 32; c < 64; c++) {
      const float* p = base + (size_t)c * HW0 + h * 64;
      for (int w = 0; w < 64; w++) acc += p[w];
    }
    acc *= (1.f / 2048.f);
  } else if (j < 160) {                       // c2: per-W, mean over ch,H
    int w = j - 96;
    for (int c = 64; c < 96; c++) {
      const float* p = base + (size_t)c * HW0 + w;
      for (int h = 0; h < 64; h++) acc += p[h * 64];
    }
    acc *= (1.f / 2048.f);
  } else if (j < 2208) {                      // c3: mean over W
    int idx = j - 160;
    int c = 96 + idx / 64, h = idx % 64;
    const float* p = base + (size_t)c * HW0 + h * 64;
    for (int w = 0; w < 64; w++) acc += p[w];
    acc *= (1.f / 64.f);
  } else if (j < 4256) {                      // c4: mean over H
    int idx = j - 2208;
    int c = 128 + idx / 64, w = idx % 64;
    const float* p = base + (size_t)c * HW0 + w;
    for (int h = 0; h < 64; h++) acc += p[h * 64];
    acc *= (1.f / 64.f);
  } else if (j < 8352) {                      // c5: mean over channels
    int idx = j - 4256;
    int h = idx / 64, w = idx % 64;
    for (int c = 160; c < 192; c++) acc += base[(size_t)c * HW0 + h * 64 + w];
    acc *= (1.f / 32.f);
  } else {                                    // c6: global mean
    for (int c = 192; c < 224; c++) {
      const float* p = base + (size_t)c * HW0;
      for (int i = 0; i < 4096; i++) acc += p[i];
    }
    acc *= (1.f / 131072.f);
  }
  css[tid] = acc;
}

__global__ void k_concat_inj(const float* __restrict__ img,
                             const float* __restrict__ labs,
                             float* __restrict__ out) {
  int tid = blockIdx.x * blockDim.x + threadIdx.x;
  if (tid >= Bsz * 512) return;
  int b = tid >> 9, j = tid & 511;
  out[tid] = (j < 256) ? img[b * 256 + j] : labs[b * 256 + j - 256];
}

__global__ void k_bcast(const float* __restrict__ src, float* __restrict__ dst,
                        int n) {
  int tid = blockIdx.x * blockDim.x + threadIdx.x;
  if (tid >= Bsz * n) return;
  dst[tid] = src[tid % n];
}

__global__ void k_pool2(const float* __restrict__ in, float* __restrict__ out,
                        int BC, int so) {
  int tid = blockIdx.x * blockDim.x + threadIdx.x;
  if (tid >= BC * so * so) return;
  int p = tid % (so * so);
  int bc = tid / (so * so);
  int y = p / so, x = p % so;
  int si = so * 2;
  const float* ib = in + (size_t)bc * si * si;
  out[tid] = 0.25f * (ib[(2 * y) * si + 2 * x] + ib[(2 * y) * si + 2 * x + 1] +
                      ib[(2 * y + 1) * si + 2 * x] +
                      ib[(2 * y + 1) * si + 2 * x + 1]);
}

__global__ void k_up2(const float* __restrict__ in, float* __restrict__ out,
                      int BC, int so) {
  int tid = blockIdx.x * blockDim.x + threadIdx.x;
  if (tid >= BC * so * so) return;
  int p = tid % (so * so);
  int bc = tid / (so * so);
  int y = p / so, x = p % so;
  int si = so / 2;
  out[tid] = in[(size_t)bc * si * si + (y >> 1) * si + (x >> 1)];
}

__global__ void k_addinto(float* __restrict__ a, const float* __restrict__ b,
                          int n) {
  int tid = blockIdx.x * blockDim.x + threadIdx.x;
  if (tid < n) a[tid] += b[tid];
}

// cat = [out(32ch) | cos(wave)(32ch) | sin(wave)(32ch)]
__global__ void k_cat96(const float* __restrict__ outF,
                        const float* __restrict__ wave,
                        float* __restrict__ cat) {
  int tid = blockIdx.x * blockDim.x + threadIdx.x;
  if (tid >= Bsz * 96 * HW0) return;
  int p = tid & 4095;
  int rest = tid >> 12;
  int c = rest % 96;
  int b = rest / 96;
  float v;
  if (c < 32)       v = outF[((size_t)b * 32 + c) * HW0 + p];
  else if (c < 64)  v = cosf(wave[((size_t)b * 32 + (c - 32)) * HW0 + p]);
  else              v = sinf(wave[((size_t)b * 32 + (c - 64)) * HW0 + p]);
  cat[tid] = v;
}

// u2 input = [l_wave(32ch) | cnew channels 32..63]
__global__ void k_u2in(const float* __restrict__ lwave,
                       const float* __restrict__ cnew,
                       float* __restrict__ dst) {
  int tid = blockIdx.x * blockDim.x + threadIdx.x;
  if (tid >= Bsz * 64 * HW0) return;
  int p = tid & 4095;
  int rest = tid >> 12;
  int c = rest & 63;
  int b = rest >> 6;
  dst[tid] = (c < 32) ? lwave[((size_t)b * 32 + c) * HW0 + p]
                      : cnew[((size_t)b * 64 + c) * HW0 + p];
}

// out[:, :32] += 0.1 * cnew[:, :32]   (cnew has 64 channels)
__global__ void k_axpy32(float* __restrict__ out,
                         const float* __restrict__ cnew) {
  int tid = blockIdx.x * blockDim.x + threadIdx.x;
  if (tid >= Bsz * 32 * HW0) return;
  int p = tid & 4095;
  int rest = tid >> 12;
  int c = rest & 31;
  int b = rest >> 5;
  out[tid] += 0.1f * cnew[((size_t)b * 64 + c) * HW0 + p];
}

// ----------------------------- host orchestration --------------------------

static inline int cdiv(long long a, int b) { return (int)((a + b - 1) / b); }

// workspace layout (float offsets)
constexpr size_t C_LABS  = 0;
constexpr size_t C_CONV1 = C_LABS + (size_t)Bsz * 256;
constexpr size_t C_CSS   = C_CONV1 + (size_t)Bsz * 224 * HW0;
constexpr size_t C_IMGL  = C_CSS + (size_t)Bsz * CSSc;
constexpr size_t C_INJ   = C_IMGL + (size_t)Bsz * 256;
constexpr size_t C_LAT   = C_INJ + (size_t)Bsz * 512;
constexpr size_t C_OUT   = C_LAT + (size_t)Bsz * 256;
constexpr size_t C_LWIN  = C_OUT + (size_t)Bsz * 32 * HW0;
constexpr size_t C_LWAVE = C_LWIN + (size_t)Bsz * 64 * HW0;
constexpr size_t C_WAVE  = C_LWAVE + (size_t)Bsz * 32 * HW0;
constexpr size_t C_KS    = C_WAVE + (size_t)Bsz * 32 * HW0;
constexpr size_t C_T1    = C_KS + (size_t)Bsz * T_CC;
constexpr size_t C_T2    = C_T1 + (size_t)Bsz * 64 * HW0;
constexpr size_t C_TS    = C_T2 + (size_t)Bsz * 64 * HW0;
constexpr size_t C_H     = C_TS + (size_t)Bsz * 64 * HW0;
constexpr size_t C_H2    = C_H + (size_t)Bsz * 64 * HW0;
constexpr size_t C_SK0   = C_H2 + (size_t)Bsz * 64 * HW0;
constexpr size_t C_SK1   = C_SK0 + (size_t)Bsz * 64 * 4096;
constexpr size_t C_SK2   = C_SK1 + (size_t)Bsz * 64 * 1024;
constexpr size_t C_SK3   = C_SK2 + (size_t)Bsz * 64 * 256;
constexpr size_t C_CAT   = C_SK3 + (size_t)Bsz * 64 * 64;
constexpr size_t C_U2IN  = C_CAT + (size_t)Bsz * 96 * HW0;
constexpr size_t C_CNEW  = C_U2IN + (size_t)Bsz * 64 * HW0;

struct DynOff { int wi, bi, wm, bm, wo, bo, ws_, bs; };
static DynOff mkoff(int fin, int fh, int fo) {
  DynOff o; int c = 0;
  o.wi = c; c += fh * fin; o.bi = c; c += fh;
  o.wm = c; c += fh * fh;  o.bm = c; c += fh;
  o.wo = c; c += fo * fh;  o.bo = c; c += fo;
  o.ws_ = c; c += fo * fin; o.bs = c; c += fo;
  return o;
}

static void launch_gemm16(const float* A, const float* W, const float* bias,
                          float* C, int K, int N, hipStream_t s) {
  int ntiles = N / 16;
  gemm16_wmma<<<cdiv(ntiles, 8), 256, 0, s>>>(A, W, bias, C, K, N);
}

static void launch_dyna(const float* ks, int kst, int woff, int boff,
                        const float* X, float* D, const float* add, int M,
                        int Kd, int HW, int relu, hipStream_t s) {
  long long threads = (long long)Bsz * (M / 16) * (HW / 16) * 32;
  int blocks = cdiv(threads, 256);
  if (Kd == 96) {
    dyna_gemm_t<96, 4096><<<blocks, 256, 0, s>>>(ks, kst, woff, boff, X, D,
                                                 add, M, relu);
  } else if (HW == 4096) {
    dyna_gemm_t<64, 4096><<<blocks, 256, 0, s>>>(ks, kst, woff, boff, X, D,
                                                 add, M, relu);
  } else if (HW == 1024) {
    dyna_gemm_t<64, 1024><<<blocks, 256, 0, s>>>(ks, kst, woff, boff, X, D,
                                                 add, M, relu);
  } else if (HW == 256) {
    dyna_gemm_t<64, 256><<<blocks, 256, 0, s>>>(ks, kst, woff, boff, X, D,
                                                add, M, relu);
  } else {
    dyna_gemm_t<64, 64><<<blocks, 256, 0, s>>>(ks, kst, woff, boff, X, D, add,
                                               M, relu);
  }
}

static void dyna_block_launch(const float* lat, const float* Wb,
                              const float* bb, int t, int fin, int fh, int fo,
                              const float* x, int HW, float* out, float* ks,
                              float* t1, float* t2, float* ts,
                              hipStream_t s) {
  launch_gemm16(lat, Wb, bb, ks, 256, t, s);          // ks = lat @ W + b
  DynOff o = mkoff(fin, fh, fo);
  launch_dyna(ks, t, o.wi, o.bi, x, t1, nullptr, fh, fin, HW, 1, s);   // relu
  launch_dyna(ks, t, o.wm, o.bm, t1, t2, nullptr, fh, fh, HW, 1, s);   // relu
  launch_dyna(ks, t, o.ws_, o.bs, x, ts, nullptr, fo, fin, HW, 0, s);  // short
  launch_dyna(ks, t, o.wo, o.bo, t2, out, ts, fo, fh, HW, 0, s);       // +s
}

static void dyna_u_launch(const float* x_in, const float* lat, const float* bw,
                          const float* bb, const float* ow, const float* ob,
                          float* out32, float* ws, hipStream_t s) {
  float* KS = ws + C_KS;
  float* T1 = ws + C_T1;
  float* T2 = ws + C_T2;
  float* TS = ws + C_TS;
  float* H  = ws + C_H;
  float* H2 = ws + C_H2;
  float* SK[4] = {ws + C_SK0, ws + C_SK1, ws + C_SK2, ws + C_SK3};
  const float* cur = x_in;
  int hw = 4096, side = 64;
  for (int i = 0; i < 4; i++) {                       // down path
    dyna_block_launch(lat, bw + (size_t)i * 256 * T_FF, bb + (size_t)i * T_FF,
                      T_FF, 64, 64, 64, cur, hw, SK[i], KS, T1, T2, TS, s);
    side /= 2; hw /= 4;
    int n = Bsz * 64 * hw;
    k_pool2<<<cdiv(n, 256), 256, 0, s>>>(SK[i], H, Bsz * 64, side);
    cur = H;
  }
  for (int i = 0; i < 4; i++) {                       // up path
    side *= 2; hw *= 4;
    int n = Bsz * 64 * hw;
    k_up2<<<cdiv(n, 256), 256, 0, s>>>(cur, H2, Bsz * 64, side);
    dyna_block_launch(lat, bw + (size_t)(4 + i) * 256 * T_FF,
                      bb + (size_t)(4 + i) * T_FF, T_FF, 64, 64, 64, H2, hw, H,
                      KS, T1, T2, TS, s);
    k_addinto<<<cdiv(n, 256), 256, 0, s>>>(H, SK[3 - i], n);
    cur = H;
  }
  dyna_block_launch(lat, ow, ob, T_FO, 64, 64, 32, H, 4096, out32, KS, T1, T2,
                    TS, s);
}

extern "C" void kernel_launch(void* const* d_in, const int* in_sizes, int n_in,
                              void* d_out, int out_size, void* d_ws,
                              size_t ws_size, hipStream_t stream) {
  (void)in_sizes; (void)n_in; (void)out_size; (void)ws_size;
  const float* x            = (const float*)d_in[0];
  const int*   labels       = (const int*)d_in[1];
  const float* W_labs       = (const float*)d_in[2];
  const float* b_labs       = (const float*)d_in[3];
  const float* conv_lat_w   = (const float*)d_in[4];
  const float* conv_lat_b   = (const float*)d_in[5];
  const float* lat_to_lat_w = (const float*)d_in[6];
  const float* lat_to_lat_b = (const float*)d_in[7];
  const float* lat_mix_w    = (const float*)d_in[8];
  const float* lat_mix_b    = (const float*)d_in[9];
  const float* conv_img_w   = (const float*)d_in[10];
  const float* conv_img_b   = (const float*)d_in[11];
  const float* wave_init    = (const float*)d_in[12];
  const float* u1_bw        = (const float*)d_in[13];
  const float* u1_bb        = (const float*)d_in[14];
  const float* u1_ow        = (const float*)d_in[15];
  const float* u1_ob        = (const float*)d_in[16];
  const float* u2_bw        = (const float*)d_in[17];
  const float* u2_bb        = (const float*)d_in[18];
  const float* u2_ow        = (const float*)d_in[19];
  const float* u2_ob        = (const float*)d_in[20];
  const float* c2c_w        = (const float*)d_in[21];
  const float* c2c_b        = (const float*)d_in[22];
  const float* out_conv_w   = (const float*)d_in[23];
  const float* out_conv_b   = (const float*)d_in[24];
  const float* out_lat_w    = (const float*)d_in[25];
  const float* out_lat_b    = (const float*)d_in[26];
  float* ws = (float*)d_ws;
  hipStream_t s = stream;

  // latent assembly
  k_gather_labs<<<cdiv(Bsz * 256, 256), 256, 0, s>>>(W_labs, b_labs, labels,
                                                     ws + C_LABS);
  k_conv3x3<<<cdiv((long long)Bsz * 224 * HW0, 256), 256, 0, s>>>(
      x, conv_lat_w, conv_lat_b, ws + C_CONV1, Bsz, 3, 224);
  k_multicut<<<cdiv(Bsz * CSSc, 256), 256, 0, s>>>(ws + C_CONV1, ws + C_CSS);
  launch_gemm16(ws + C_CSS, lat_to_lat_w, lat_to_lat_b, ws + C_IMGL, CSSc, 256,
                s);
  k_concat_inj<<<cdiv(Bsz * 512, 256), 256, 0, s>>>(ws + C_IMGL, ws + C_LABS,
                                                    ws + C_INJ);
  launch_gemm16(ws + C_INJ, lat_mix_w, lat_mix_b, ws + C_LAT, 512, 256, s);

  // image path init
  k_conv3x3<<<cdiv((long long)Bsz * 32 * HW0, 256), 256, 0, s>>>(
      x, conv_img_w, conv_img_b, ws + C_OUT, Bsz, 3, 32);
  k_bcast<<<cdiv((long long)Bsz * 64 * HW0, 256), 256, 0, s>>>(
      wave_init, ws + C_LWIN, 64 * HW0);

  // u1
  dyna_u_launch(ws + C_LWIN, ws + C_LAT, u1_bw, u1_bb, u1_ow, u1_ob,
                ws + C_LWAVE, ws, s);

  const float* wavep = ws + C_LWAVE;
  for (int c = 0; c < 8; c++) {
    k_cat96<<<cdiv((long long)Bsz * 96 * HW0, 256), 256, 0, s>>>(
        ws + C_OUT, wavep, ws + C_CAT);
    dyna_block_launch(ws + C_LAT, c2c_w, c2c_b, T_CC, 96, 64, 64, ws + C_CAT,
                      HW0, ws + C_CNEW, ws + C_KS, ws + C_T1, ws + C_T2,
                      ws + C_TS, s);
    if (c < 7) {
      k_u2in<<<cdiv((long long)Bsz * 64 * HW0, 256), 256, 0, s>>>(
          ws + C_LWAVE, ws + C_CNEW, ws + C_U2IN);
      dyna_u_launch(ws + C_U2IN, ws + C_LAT, u2_bw, u2_bb, u2_ow, u2_ob,
                    ws + C_WAVE, ws, s);
      wavep = ws + C_WAVE;
    }
    k_axpy32<<<cdiv((long long)Bsz * 32 * HW0, 256), 256, 0, s>>>(ws + C_OUT,
                                                                  ws + C_CNEW);
  }

  // readout
  k_conv3x3<<<cdiv((long long)Bsz * 224 * HW0, 256), 256, 0, s>>>(
      ws + C_OUT, out_conv_w, out_conv_b, ws + C_CONV1, Bsz, 32, 224);
  k_multicut<<<cdiv(Bsz * CSSc, 256), 256, 0, s>>>(ws + C_CONV1, ws + C_CSS);
  launch_gemm16(ws + C_CSS, out_lat_w, out_lat_b, (float*)d_out, CSSc, 256, s);
}